// SPRGNN_88648124991069
// MI455X (gfx1250) — compile-verified
//
#include <hip/hip_runtime.h>
#include <hip/hip_bf16.h>
#include <math.h>

typedef float v2f __attribute__((ext_vector_type(2)));
typedef float v8f __attribute__((ext_vector_type(8)));

#define FIXSCALE 4294967296.0   // 2^32 Q32 fixed point
#define INVFIX   (1.0 / 4294967296.0)

// ---------------------------------------------------------------------------
// Node feature build: h0[n] = concat(shape[x0], color[x1], pos[x2])  -> [N,80]
// ---------------------------------------------------------------------------
__global__ void sprgnn_emb_kernel(const int* __restrict__ x,
                                  const float* __restrict__ se,
                                  const float* __restrict__ ce,
                                  const float* __restrict__ pe,
                                  float* __restrict__ h, int N) {
    int gid = blockIdx.x * blockDim.x + threadIdx.x;
    if (gid >= N * 80) return;
    int n = gid / 80, j = gid - n * 80;
    float v;
    if (j < 32)      v = se[x[n * 3 + 0] * 32 + j];
    else if (j < 64) v = ce[x[n * 3 + 1] * 32 + (j - 32)];
    else             v = pe[x[n * 3 + 2] * 16 + (j - 64)];
    h[gid] = v;
}

// ---------------------------------------------------------------------------
// In-degree (exact integer atomics -> deterministic)
// ---------------------------------------------------------------------------
__global__ void sprgnn_deg_kernel(const int* __restrict__ dst,
                                  unsigned* __restrict__ deg, int E) {
    int e = blockIdx.x * blockDim.x + threadIdx.x;
    if (e < E) atomicAdd(&deg[dst[e]], 1u);
}

// ---------------------------------------------------------------------------
// Fused dual GEMM via V_WMMA_F32_16X16X4_F32:
//   T[:,0:64]   = A @ Wl    (to be aggregated over edges)
//   T[:,64:128] = A @ Wr    (self/root term)
// A: [M,K] row-major, Wl/Wr: [K,64] row-major, T: [M,128].
// Block = 256 thr = 8 waves; block covers 64 rows x 128 cols.
// Wave w owns 16 output cols; loops over 4 M-tiles of 16 rows.
// A-tile staged in LDS (stride K+2: conflict-free, 8B aligned).
// ---------------------------------------------------------------------------
template <int K>
__global__ __launch_bounds__(256) void sprgnn_gemm_wmma(
    const float* __restrict__ A, const float* __restrict__ Wl,
    const float* __restrict__ Wr, float* __restrict__ T, int M) {
    constexpr int LDK = K + 2;
    __shared__ float As[64 * LDK];

    const int tid = threadIdx.x;
    const int block_row = blockIdx.x * 64;

    // cooperative zero-padded load of 64 x K A-tile
    for (int i = tid; i < 64 * K; i += 256) {
        int r = i / K, c = i - r * K;
        int gr = block_row + r;
        As[r * LDK + c] = (gr < M) ? A[(long long)gr * K + c] : 0.0f;
    }
    __syncthreads();

    const int w     = tid >> 5;        // wave 0..7
    const int lane  = tid & 31;
    const int n     = lane & 15;       // N col / A row within tile
    const int khalf = lane >> 4;       // 0: K pair {0,1}; 1: K pair {2,3}
    const float* Wp = (w < 4) ? Wl : Wr;
    const int colbase = (w < 4) ? (w * 16) : ((w - 4) * 16);
    const int outcol  = (w < 4) ? (w * 16 + n) : (64 + (w - 4) * 16 + n);

    for (int mt = 0; mt < 4; ++mt) {
        v8f acc = {0.f, 0.f, 0.f, 0.f, 0.f, 0.f, 0.f, 0.f};
        const float* arow = &As[(mt * 16 + n) * LDK];
#pragma unroll
        for (int kk = 0; kk < K / 4; ++kk) {
            const int kb = kk * 4 + khalf * 2;
            v2f a, b;
            a.x = arow[kb];
            a.y = arow[kb + 1];
            b.x = Wp[kb * 64 + colbase + n];
            b.y = Wp[(kb + 1) * 64 + colbase + n];
            acc = __builtin_amdgcn_wmma_f32_16x16x4_f32(
                false, a, false, b, (short)0, acc, false, false);
        }
#pragma unroll
        for (int v = 0; v < 8; ++v) {
            int r = block_row + mt * 16 + v + 8 * khalf;
            if (r < M) T[(long long)r * 128 + outcol] = acc[v];
        }
    }
}

// ---------------------------------------------------------------------------
// Edge scatter-add (Q32 fixed point -> deterministic):
//   agg[dst[e]][j] += T[src[e]][j],  j in [0,64)
// ---------------------------------------------------------------------------
__global__ void sprgnn_scatter_kernel(const float* __restrict__ T,
                                      const int* __restrict__ src,
                                      const int* __restrict__ dst,
                                      unsigned long long* __restrict__ agg,
                                      int nwork) {
    int gid = blockIdx.x * blockDim.x + threadIdx.x;
    if (gid >= nwork) return;
    int e = gid >> 6, j = gid & 63;
    float v = T[(long long)src[e] * 128 + j];
    long long q = llrint((double)v * FIXSCALE);
    atomicAdd(&agg[(long long)dst[e] * 64 + j], (unsigned long long)q);
}

// ---------------------------------------------------------------------------
// Combine: H[n][j] = relu(agg[n][j]/max(deg,1) + b[j] + T[n][64+j])
// ---------------------------------------------------------------------------
__global__ void sprgnn_combine_kernel(const unsigned long long* __restrict__ agg,
                                      const unsigned* __restrict__ deg,
                                      const float* __restrict__ bias,
                                      const float* __restrict__ T,
                                      float* __restrict__ H, int N) {
    int gid = blockIdx.x * blockDim.x + threadIdx.x;
    if (gid >= N * 64) return;
    int n = gid >> 6, j = gid & 63;
    float a = (float)((double)(long long)agg[gid] * INVFIX);
    float dinv = 1.0f / fmaxf((float)deg[n], 1.0f);
    float v = a * dinv + bias[j] + T[(long long)n * 128 + 64 + j];
    H[gid] = fmaxf(v, 0.0f);
}

// ---------------------------------------------------------------------------
// Graph mean-pool accumulation (Q32 + integer count -> deterministic)
// ---------------------------------------------------------------------------
__global__ void sprgnn_pool_kernel(const float* __restrict__ H,
                                   const int* __restrict__ batch,
                                   unsigned long long* __restrict__ pool,
                                   unsigned* __restrict__ cnt, int N) {
    int gid = blockIdx.x * blockDim.x + threadIdx.x;
    if (gid >= N * 64) return;
    int n = gid >> 6, j = gid & 63;
    int g = batch[n];
    long long q = llrint((double)H[gid] * FIXSCALE);
    atomicAdd(&pool[(long long)g * 64 + j], (unsigned long long)q);
    if (j == 0) atomicAdd(&cnt[g], 1u);
}

// ---------------------------------------------------------------------------
// Classifier: out[g][c] = (pool[g]/max(cnt,1)) . Wc[:,c] + bc[c]
// ---------------------------------------------------------------------------
__global__ void sprgnn_cls_kernel(const unsigned long long* __restrict__ pool,
                                  const unsigned* __restrict__ cnt,
                                  const float* __restrict__ Wc,
                                  const float* __restrict__ bc,
                                  float* __restrict__ out, int G) {
    int gid = blockIdx.x * blockDim.x + threadIdx.x;
    int g = gid >> 4, c = gid & 15;
    if (g >= G || c >= 10) return;
    float inv = 1.0f / fmaxf((float)cnt[g], 1.0f);
    float acc = bc[c];
#pragma unroll 8
    for (int j = 0; j < 64; ++j) {
        float hv = (float)((double)(long long)pool[(long long)g * 64 + j] * INVFIX);
        acc = fmaf(hv * inv, Wc[j * 10 + c], acc);
    }
    out[g * 10 + c] = acc;
}

// ---------------------------------------------------------------------------
extern "C" void kernel_launch(void* const* d_in, const int* in_sizes, int n_in,
                              void* d_out, int out_size, void* d_ws, size_t ws_size,
                              hipStream_t stream) {
    const int* x      = (const int*)d_in[0];
    const int* ei     = (const int*)d_in[1];
    const int* batch  = (const int*)d_in[2];
    const float* se   = (const float*)d_in[3];
    const float* ce   = (const float*)d_in[4];
    const float* pe   = (const float*)d_in[5];
    const float* W1l  = (const float*)d_in[6];
    const float* b1   = (const float*)d_in[7];
    const float* W1r  = (const float*)d_in[8];
    const float* W2l  = (const float*)d_in[9];
    const float* b2   = (const float*)d_in[10];
    const float* W2r  = (const float*)d_in[11];
    const float* Wc   = (const float*)d_in[12];
    const float* bc   = (const float*)d_in[13];
    float* out = (float*)d_out;

    const int N = in_sizes[2];          // 100000 (batch length)
    const int E = in_sizes[1] / 2;      // 1600000
    const int G = out_size / 10;        // 5000
    const int* src = ei;
    const int* dst = ei + E;

    // workspace layout (256B aligned slices)
    char* ws = (char*)d_ws;
    size_t off = 0;
    auto take = [&](size_t bytes) -> char* {
        char* p = ws + off;
        off += (bytes + 255) & ~(size_t)255;
        return p;
    };
    float* hbuf = (float*)take((size_t)N * 80 * 4);                 // h0 / h1 / h2
    float* tbuf = (float*)take((size_t)N * 128 * 4);                // dual-GEMM output
    unsigned long long* agg = (unsigned long long*)take((size_t)N * 64 * 8);
    unsigned* deg = (unsigned*)take((size_t)N * 4);
    unsigned long long* pool = (unsigned long long*)take((size_t)G * 64 * 8);
    unsigned* cnt = (unsigned*)take((size_t)G * 4);
    (void)ws_size;

    const int TPB = 256;
    auto blocks = [](long long w, int t) { return (unsigned)((w + t - 1) / t); };

    // zero accumulators (fresh every launch; capture-safe)
    hipMemsetAsync(deg, 0, (size_t)N * 4, stream);
    hipMemsetAsync(agg, 0, (size_t)N * 64 * 8, stream);
    hipMemsetAsync(pool, 0, (size_t)G * 64 * 8, stream);
    hipMemsetAsync(cnt, 0, (size_t)G * 4, stream);

    // h0 = embeddings, deg = in-degree
    sprgnn_emb_kernel<<<blocks((long long)N * 80, TPB), TPB, 0, stream>>>(x, se, ce, pe, hbuf, N);
    sprgnn_deg_kernel<<<blocks(E, TPB), TPB, 0, stream>>>(dst, deg, E);

    // layer 1: transform-then-aggregate
    sprgnn_gemm_wmma<80><<<blocks(((long long)N + 63) / 64 * 64, 64) , TPB, 0, stream>>>(hbuf, W1l, W1r, tbuf, N);
    sprgnn_scatter_kernel<<<blocks((long long)E * 64, TPB), TPB, 0, stream>>>(tbuf, src, dst, agg, E * 64);
    sprgnn_combine_kernel<<<blocks((long long)N * 64, TPB), TPB, 0, stream>>>(agg, deg, b1, tbuf, hbuf, N);

    // layer 2
    hipMemsetAsync(agg, 0, (size_t)N * 64 * 8, stream);
    sprgnn_gemm_wmma<64><<<blocks(((long long)N + 63) / 64 * 64, 64), TPB, 0, stream>>>(hbuf, W2l, W2r, tbuf, N);
    sprgnn_scatter_kernel<<<blocks((long long)E * 64, TPB), TPB, 0, stream>>>(tbuf, src, dst, agg, E * 64);
    sprgnn_combine_kernel<<<blocks((long long)N * 64, TPB), TPB, 0, stream>>>(agg, deg, b2, tbuf, hbuf, N);

    // mean pool + classifier
    sprgnn_pool_kernel<<<blocks((long long)N * 64, TPB), TPB, 0, stream>>>(hbuf, batch, pool, cnt, N);
    sprgnn_cls_kernel<<<blocks((long long)G * 16, TPB), TPB, 0, stream>>>(pool, cnt, Wc, bc, out, G);
}